// Pyramidal_20461224198253
// MI455X (gfx1250) — compile-verified
//
#include <hip/hip_runtime.h>

// Problem constants (match reference globals)
#define Bdim 1024
#define Pdim 1024
#define Hdim 4096
#define Ndim 32
#define Ddim 512

typedef __attribute__((ext_vector_type(2))) float v2f;
typedef __attribute__((ext_vector_type(8))) float v8f;

// ---------------- proximal GEMM tiling ----------------
#define KC    64   // K chunk staged in LDS
#define LDSTR 68   // KC + 4 pad: conflict-free column reads, 16B-aligned rows

// ---------------- distal GEMM tiling ------------------
#define KC2   32   // A-chunk K width
#define ASTR  36   // KC2 + 4 pad
#define BSTR  516  // Ddim + 4 pad (resident B panel row stride)

// ---------------------------------------------------------------------------
// Kernel 1: distal GEMM with B-panel resident in LDS.
// Each block owns 16 hn-columns (half of one hidden unit's 32 dendrites) for
// the FULL K=512 (33KB LDS) -> distal weights stream from HBM exactly once
// (256 MB total instead of 16 GB). The block loops over all 1024 batch rows
// in 8 m-iters (8 waves x 16 rows), staging A chunks (L2-resident, 2 MB).
// Emits per-(b, h, half) signed abs-max partials; combine happens later.
// ---------------------------------------------------------------------------
__global__ __launch_bounds__(256) void k_distal_partial(
    const float* __restrict__ Xd, const float* __restrict__ distal,
    float* __restrict__ part)   // (B, H, 2) signed abs-max partials
{
    __shared__ float lB[16 * BSTR];    // 33,024 B  (resident panel)
    __shared__ float lA[128 * ASTR];   // 18,432 B  (chunk staging / epilogue)

    const int tid  = threadIdx.x;
    const int lane = tid & 31;
    const int wave = tid >> 5;
    const int hn0  = blockIdx.x << 4;       // panel base in hn
    const int h    = blockIdx.x >> 1;       // hidden unit
    const int half = blockIdx.x & 1;        // dendrites [0..15] or [16..31]

    // Load resident B panel: 16 rows x 512 cols (batched loads -> stores)
    {
        float4 t[8];
        #pragma unroll
        for (int j = 0; j < 8; ++j) {
            const int f   = tid + j * 256;      // float4 idx 0..2047
            const int row = f >> 7;             // 0..15
            const int col = (f & 127) << 2;     // 0..508
            t[j] = *reinterpret_cast<const float4*>(
                &distal[(size_t)(hn0 + row) * Ddim + col]);
        }
        #pragma unroll
        for (int j = 0; j < 8; ++j) {
            const int f   = tid + j * 256;
            const int row = f >> 7;
            const int col = (f & 127) << 2;
            *reinterpret_cast<float4*>(&lB[row * BSTR + col]) = t[j];
        }
    }

    const int mrow = lane & 15;             // row/col within 16-lane group
    const int kb   = (lane >> 4) << 1;      // lanes 16-31 hold K+2 (ISA layout)
    const int arow = tid >> 3;              // A prefetch row 0..31
    const int acol = (tid & 7) << 2;        // A staging col 0..28

    for (int mi = 0; mi < 8; ++mi) {
        const int mbase = mi << 7;          // 128 batch rows per m-iter
        v8f acc0 = {}, acc1 = {};
        for (int kc = 0; kc < Ddim; kc += KC2) {
            __syncthreads();  // prior LDS readers done before restage
            // Stage A chunk: 128 x 32 (batched: 4 loads then 4 stores)
            float4 t[4];
            #pragma unroll
            for (int j = 0; j < 4; ++j) {
                const int f   = tid + j * 256;
                const int row = f >> 3;          // 0..127
                const int col = (f & 7) << 2;    // 0..28
                t[j] = *reinterpret_cast<const float4*>(
                    &Xd[(size_t)(mbase + row) * Ddim + kc + col]);
            }
            #pragma unroll
            for (int j = 0; j < 4; ++j) {
                const int f   = tid + j * 256;
                const int row = f >> 3;
                const int col = (f & 7) << 2;
                *reinterpret_cast<float4*>(&lA[row * ASTR + col]) = t[j];
            }
            // Prefetch next chunk into cache (global_prefetch_b8)
            if (kc + KC2 < Ddim)
                __builtin_prefetch(&Xd[(size_t)(mbase + arow) * Ddim +
                                       kc + KC2 + acol], 0, 3);
            __syncthreads();
            const float* Ar = &lA[((wave << 4) + mrow) * ASTR];
            const float* Br = &lB[mrow * BSTR + kc];
            #pragma unroll
            for (int kk = 0; kk < KC2; kk += 8) {
                v2f a, b;
                a.x = Ar[kk + kb];     a.y = Ar[kk + kb + 1];
                b.x = Br[kk + kb];     b.y = Br[kk + kb + 1];
                acc0 = __builtin_amdgcn_wmma_f32_16x16x4_f32(
                    false, a, false, b, (short)0, acc0, false, false);
                a.x = Ar[kk + 4 + kb]; a.y = Ar[kk + 4 + kb + 1];
                b.x = Br[kk + 4 + kb]; b.y = Br[kk + 4 + kb + 1];
                acc1 = __builtin_amdgcn_wmma_f32_16x16x4_f32(
                    false, a, false, b, (short)0, acc1, false, false);
            }
        }
        // Epilogue: per batch row, abs-argmax over this panel's 16 dendrites
        __syncthreads();                   // WMMA LDS readers done
        float* sres = lA;                  // reuse: 128 rows x 16 cols, stride 17
        const int rb = (lane >> 4) << 3;   // C/D layout: lanes 16-31 hold M+8
        #pragma unroll
        for (int r = 0; r < 8; ++r)
            sres[((wave << 4) + rb + r) * 17 + mrow] = acc0[r] + acc1[r];
        __syncthreads();
        if (tid < 128) {
            const float* v = &sres[tid * 17];
            float best  = v[0];
            float besta = fabsf(best);
            #pragma unroll
            for (int j = 1; j < 16; ++j) {
                const float x = v[j], ax = fabsf(x);
                if (ax > besta) { besta = ax; best = x; }  // first-max ties
            }
            part[((size_t)(mbase + tid) * Hdim + h) * 2 + half] = best;
        }
        // next m-iter's first __syncthreads() protects lA reuse
    }
}

// ---------------------------------------------------------------------------
// Kernel 2: proximal GEMM + bias -> prox (B x H).
// Block = 8 waves, tile 16M x 128N, K chunks of 64 staged in LDS.
// W (16 MB) becomes L2-resident after one N-sweep, so traffic is fine.
// ---------------------------------------------------------------------------
__global__ __launch_bounds__(256) void k_prox_gemm(
    const float* __restrict__ X, const float* __restrict__ W,
    const float* __restrict__ bias, float* __restrict__ prox)
{
    __shared__ float lA[16 * LDSTR];
    __shared__ float lB[128 * LDSTR];
    const int tid   = threadIdx.x;
    const int lane  = tid & 31;
    const int wave  = tid >> 5;
    const int n0blk = blockIdx.x << 7;
    const int m0    = blockIdx.y << 4;
    const int mrow  = lane & 15;
    const int kb    = (lane >> 4) << 1;
    const int nl    = (wave << 4) + mrow;

    v8f acc0 = {}, acc1 = {};
    for (int k0 = 0; k0 < Pdim; k0 += KC) {
        __syncthreads();
        // Batched staging: all global loads first, then LDS stores
        float4 ta, tb[8];
        {
            const int row = tid >> 4;
            const int col = (tid & 15) << 2;
            ta = *reinterpret_cast<const float4*>(
                &X[(size_t)(m0 + row) * Pdim + k0 + col]);
        }
        #pragma unroll
        for (int j = 0; j < 8; ++j) {
            const int f   = tid + j * 256;
            const int row = f >> 4;
            const int col = (f & 15) << 2;
            tb[j] = *reinterpret_cast<const float4*>(
                &W[(size_t)(n0blk + row) * Pdim + k0 + col]);
        }
        {
            const int row = tid >> 4;
            const int col = (tid & 15) << 2;
            *reinterpret_cast<float4*>(&lA[row * LDSTR + col]) = ta;
        }
        #pragma unroll
        for (int j = 0; j < 8; ++j) {
            const int f   = tid + j * 256;
            const int row = f >> 4;
            const int col = (f & 15) << 2;
            *reinterpret_cast<float4*>(&lB[row * LDSTR + col]) = tb[j];
        }
        __syncthreads();
        const float* Ar = &lA[mrow * LDSTR];
        const float* Br = &lB[nl * LDSTR];
        #pragma unroll
        for (int kk = 0; kk < KC; kk += 8) {
            v2f a, b;
            a.x = Ar[kk + kb];     a.y = Ar[kk + kb + 1];
            b.x = Br[kk + kb];     b.y = Br[kk + kb + 1];
            acc0 = __builtin_amdgcn_wmma_f32_16x16x4_f32(
                false, a, false, b, (short)0, acc0, false, false);
            a.x = Ar[kk + 4 + kb]; a.y = Ar[kk + 4 + kb + 1];
            b.x = Br[kk + 4 + kb]; b.y = Br[kk + 4 + kb + 1];
            acc1 = __builtin_amdgcn_wmma_f32_16x16x4_f32(
                false, a, false, b, (short)0, acc1, false, false);
        }
    }
    const int col   = n0blk + (wave << 4) + mrow;
    const int rbase = m0 + ((lane >> 4) << 3);
    const float bv  = bias[col];
    #pragma unroll
    for (int r = 0; r < 8; ++r)
        prox[(size_t)(rbase + r) * Hdim + col] = acc0[r] + acc1[r] + bv;
}

// ---------------------------------------------------------------------------
// Kernel 3: combine dendrite-half partials -> sigmoid gate, result =
// prox*gate, exact deterministic per-row top-k mask.
// ---------------------------------------------------------------------------
__device__ __forceinline__ unsigned f32_sortable(float f) {
    const unsigned u = __float_as_uint(f);
    return (u & 0x80000000u) ? ~u : (u | 0x80000000u);
}

__global__ __launch_bounds__(256) void k_gate_topk(
    const float* __restrict__ prox, const float* __restrict__ part,
    const int* __restrict__ kptr, float* __restrict__ out)
{
    __shared__ float    vals[Hdim];
    __shared__ unsigned uu[Hdim];
    __shared__ unsigned cnt[256];
    __shared__ unsigned pref[256];

    const int b   = blockIdx.x;
    const int tid = threadIdx.x;
    const unsigned kk = (unsigned)kptr[0];

    for (int i = tid; i < Hdim; i += 256) {
        const size_t idx = (size_t)b * Hdim + i;
        const float p0 = part[idx * 2 + 0];     // dendrites 0..15 winner
        const float p1 = part[idx * 2 + 1];     // dendrites 16..31 winner
        // lower half wins ties -> matches argmax first-index rule
        const float dm = (fabsf(p1) > fabsf(p0)) ? p1 : p0;
        const float g  = 1.0f / (1.0f + expf(-dm));
        const float r  = prox[idx] * g;
        vals[i] = r;
        uu[i]   = f32_sortable(r);
    }
    __syncthreads();

    const int i0 = tid * (Hdim / 256);   // contiguous chunk of 16

    // 32-step bisection: largest thr with count(u >= thr) >= k == k-th value
    unsigned thr = 0u;
    for (int bit = 31; bit >= 0; --bit) {
        const unsigned cand = thr | (1u << bit);
        unsigned c = 0;
        #pragma unroll
        for (int i = 0; i < Hdim / 256; ++i) c += (uu[i0 + i] >= cand);
        cnt[tid] = c;
        __syncthreads();
        for (int s = 128; s > 0; s >>= 1) {
            if (tid < s) cnt[tid] += cnt[tid + s];
            __syncthreads();
        }
        const unsigned total = cnt[0];
        __syncthreads();
        if (total >= kk) thr = cand;     // uniform across block
    }

    unsigned cg = 0, ce = 0;
    #pragma unroll
    for (int i = 0; i < Hdim / 256; ++i) {
        const unsigned u = uu[i0 + i];
        cg += (u > thr);
        ce += (u == thr);
    }
    cnt[tid] = cg;
    __syncthreads();
    for (int s = 128; s > 0; s >>= 1) {
        if (tid < s) cnt[tid] += cnt[tid + s];
        __syncthreads();
    }
    const unsigned totalg = cnt[0];
    __syncthreads();
    const unsigned needeq = (kk > totalg) ? (kk - totalg) : 0u;

    cnt[tid] = ce;
    __syncthreads();
    if (tid == 0) {
        unsigned run = 0;
        for (int t = 0; t < 256; ++t) { pref[t] = run; run += cnt[t]; }
    }
    __syncthreads();
    unsigned myeq = pref[tid];

    #pragma unroll
    for (int i = 0; i < Hdim / 256; ++i) {
        const unsigned u = uu[i0 + i];
        float o = 0.0f;
        if (u > thr)            o = vals[i0 + i];
        else if (u == thr) { if (myeq < needeq) o = vals[i0 + i]; ++myeq; }
        out[(size_t)b * Hdim + i0 + i] = o;
    }
}

// ---------------------------------------------------------------------------
extern "C" void kernel_launch(void* const* d_in, const int* in_sizes, int n_in,
                              void* d_out, int out_size, void* d_ws, size_t ws_size,
                              hipStream_t stream) {
    const float* prox_in = (const float*)d_in[0];   // (B,P)
    const float* dist_in = (const float*)d_in[1];   // (B,D)
    const float* W       = (const float*)d_in[2];   // (H,P)
    const float* bvec    = (const float*)d_in[3];   // (H,)
    const float* distal  = (const float*)d_in[4];   // (H,N,D)
    const int*   kptr    = (const int*)d_in[5];     // scalar k

    float* prox = (float*)d_ws;                          // B*H      (16 MB)
    float* part = prox + (size_t)Bdim * Hdim;            // B*H*2    (32 MB)
    float* outp = (float*)d_out;

    const dim3 blk(256);
    // distal GEMM, weights streamed from HBM exactly once (~256 MB)
    k_distal_partial<<<dim3((Hdim * Ndim) / 16), blk, 0, stream>>>(
        dist_in, distal, part);
    // proximal GEMM + bias
    k_prox_gemm<<<dim3(Hdim / 128, Bdim / 16), blk, 0, stream>>>(
        prox_in, W, bvec, prox);
    // gate combine + multiply + exact top-k mask
    k_gate_topk<<<dim3(Bdim), blk, 0, stream>>>(prox, part, kptr, outp);
}